// Consistent_loss_up_2_25288767439315
// MI455X (gfx1250) — compile-verified
//
#include <hip/hip_runtime.h>

// Problem constants (from reference): B=128, C=1, H=W=256.
#define BATCH   128
#define HGT     256
#define WID     256
#define TILE_J  32                          // j-columns per workgroup
#define THREADS 256
#define TILES_PER_B (WID / TILE_J)          // 8
#define NBLOCKS (BATCH * TILES_PER_B)       // 1024
#define MAP_ELEMS (TILE_J * WID)            // 8192 floats per map (32 KB)
#define CHUNKS (MAP_ELEMS / 4 / THREADS)    // 8 float4 chunks per map per thread

typedef float v4f __attribute__((ext_vector_type(4)));
typedef int   v4i __attribute__((ext_vector_type(4)));
typedef __attribute__((address_space(1))) v4i* gptr_v4i;  // global src for async DMA
typedef __attribute__((address_space(3))) v4i* lptr_v4i;  // LDS dst for async DMA

#if defined(__has_builtin)
# if __has_builtin(__builtin_amdgcn_global_load_async_to_lds_b128)
#  define HAVE_ASYNC_LDS 1
# endif
# if __has_builtin(__builtin_amdgcn_s_wait_asynccnt)
#  define HAVE_WAIT_ASYNC 1
# endif
#endif

#if defined(HAVE_ASYNC_LDS)
__device__ __forceinline__ void wait_async_zero() {
#if defined(HAVE_WAIT_ASYNC)
  __builtin_amdgcn_s_wait_asynccnt(0);
#else
  asm volatile("s_wait_asynccnt 0" ::: "memory");
#endif
}
#endif

__global__ void __launch_bounds__(THREADS)
consistency_scatter_kernel(const float* __restrict__ up,
                           const float* __restrict__ left,
                           const float* __restrict__ right,
                           float* __restrict__ partials)
{
  // 129 KB static LDS — fits the 320 KB/WGP budget of gfx1250.
  __shared__ float mapL[MAP_ELEMS];
  __shared__ float mapR[MAP_ELEMS];
  __shared__ float stgL[MAP_ELEMS];
  __shared__ float stgR[MAP_ELEMS];
  __shared__ float red[THREADS];

  const int t   = threadIdx.x;
  const int blk = blockIdx.x;
  const int k   = blk / TILES_PER_B;
  const int j0  = (blk % TILES_PER_B) * TILE_J;

  // left/right tile [k, 0, j0:j0+TILE_J, :] is a contiguous run of MAP_ELEMS floats.
  const size_t tile_base = ((size_t)k * WID + j0) * (size_t)WID;

  // ---- Phase 0: kick off async DMA of left/right tiles into LDS (overlaps scatter).
  //      Each thread issues exactly the chunks it will consume, so its own
  //      ASYNCcnt==0 wait is sufficient (no cross-wave dependency).
#if defined(HAVE_ASYNC_LDS)
  {
    const v4f* gl = (const v4f*)(left  + tile_base);
    const v4f* gr = (const v4f*)(right + tile_base);
    v4f* sl = (v4f*)stgL;
    v4f* sr = (v4f*)stgR;
#pragma unroll
    for (int it = 0; it < CHUNKS; ++it) {
      const int m4 = it * THREADS + t;
      __builtin_amdgcn_global_load_async_to_lds_b128(
          (gptr_v4i)(gl + m4), (lptr_v4i)(sl + m4), 0, 0);
      __builtin_amdgcn_global_load_async_to_lds_b128(
          (gptr_v4i)(gr + m4), (lptr_v4i)(sr + m4), 0, 0);
    }
  }
#endif

  // ---- Phase 1: zero the scatter maps.
  for (int m = t; m < MAP_ELEMS; m += THREADS) {
    mapL[m] = 0.0f;
    mapR[m] = 0.0f;
  }
  __syncthreads();

  // ---- Phase 2: scatter-max over rows i for this j-tile.
  // Lanes 0..31 of a wave share one row i (contiguous 128B up loads); 8 rows/iter.
  const int jr   = t & (TILE_J - 1);
  const int ioff = t >> 5;                      // 0..7
  const float* upcol = up + (size_t)k * HGT * WID + (size_t)j0 + jr;
  for (int ib = 0; ib < HGT; ib += (THREADS / TILE_J)) {
    const int i = ib + ioff;
    const float v = upcol[(size_t)i * WID];
    if (i + (THREADS / TILE_J) < HGT)
      __builtin_prefetch(&upcol[(size_t)(i + (THREADS / TILE_J)) * WID], 0, 3);
    if (v >= 0.0235f) {
      // round-half-even (matches jnp.round); unfused mul+add matches XLA f32 math.
      int bin = (int)rintf(__fadd_rn(__fmul_rn(v, 50.0f), 110.0f));
      bin = bin < 0 ? 0 : (bin > WID - 1 ? WID - 1 : bin);
      if (i > 128) {
        const float val = ((float)i - 128.0f) / 60.0f;   // >= 0
        atomicMax(reinterpret_cast<unsigned int*>(&mapR[jr * WID + bin]),
                  __float_as_uint(val));                 // ds_max_u32 == float max (vals >= 0)
      } else {
        const float val = (128.0f - (float)i) / 60.0f;   // >= 0
        atomicMax(reinterpret_cast<unsigned int*>(&mapL[jr * WID + bin]),
                  __float_as_uint(val));
      }
    }
  }
  __syncthreads();

  // ---- Phase 3: masked |diff| against left/right, accumulate.
#if defined(HAVE_ASYNC_LDS)
  wait_async_zero();
  const v4f* vl = (const v4f*)stgL;
  const v4f* vr = (const v4f*)stgR;
#else
  const v4f* vl = (const v4f*)(left  + tile_base);
  const v4f* vr = (const v4f*)(right + tile_base);
#endif
  const v4f* vmapL = (const v4f*)mapL;
  const v4f* vmapR = (const v4f*)mapR;
  float acc = 0.0f;
#pragma unroll
  for (int it = 0; it < CHUNKS; ++it) {
    const int m4 = it * THREADS + t;
    const v4f ul4 = vmapL[m4];
    const v4f ur4 = vmapR[m4];
    const v4f l4  = vl[m4];
    const v4f r4  = vr[m4];
#pragma unroll
    for (int e = 0; e < 4; ++e) {
      const float ul = ul4[e];
      const float dl = fabsf(ul - l4[e]);
      acc += (dl < 0.2f && ul != 0.0f) ? dl : 0.0f;
      const float ur = ur4[e];
      const float dr = fabsf(ur - r4[e]);
      acc += (dr < 0.2f && ur != 0.0f) ? dr : 0.0f;
    }
  }

  // ---- Phase 4: deterministic block tree reduction.
  red[t] = acc;
  __syncthreads();
  for (int s = THREADS / 2; s > 0; s >>= 1) {
    if (t < s) red[t] += red[t + s];
    __syncthreads();
  }
  if (t == 0) partials[blk] = red[0];
}

__global__ void __launch_bounds__(THREADS)
final_reduce_kernel(const float* __restrict__ partials, float* __restrict__ out)
{
  __shared__ double sred[THREADS];
  double a = 0.0;
  for (int m = threadIdx.x; m < NBLOCKS; m += THREADS) a += (double)partials[m];
  sred[threadIdx.x] = a;
  __syncthreads();
  for (int s = THREADS / 2; s > 0; s >>= 1) {
    if (threadIdx.x < s) sred[threadIdx.x] += sred[threadIdx.x + s];
    __syncthreads();
  }
  if (threadIdx.x == 0) {
    const double n = (double)BATCH * (double)WID * (double)WID;  // same N for both means
    out[0] = (float)(sred[0] / n);
  }
}

extern "C" void kernel_launch(void* const* d_in, const int* in_sizes, int n_in,
                              void* d_out, int out_size, void* d_ws, size_t ws_size,
                              hipStream_t stream) {
  (void)in_sizes; (void)n_in; (void)out_size; (void)ws_size;
  const float* up    = (const float*)d_in[0];
  const float* left  = (const float*)d_in[1];
  const float* right = (const float*)d_in[2];
  float* partials = (float*)d_ws;   // NBLOCKS floats = 4 KB scratch

  consistency_scatter_kernel<<<dim3(NBLOCKS), dim3(THREADS), 0, stream>>>(
      up, left, right, partials);
  final_reduce_kernel<<<dim3(1), dim3(THREADS), 0, stream>>>(
      partials, (float*)d_out);
}